// MoeConv2d_35845797053227
// MI455X (gfx1250) — compile-verified
//
#include <hip/hip_runtime.h>

typedef __attribute__((ext_vector_type(2))) float v2f;
typedef __attribute__((ext_vector_type(4))) float v4f;
typedef __attribute__((ext_vector_type(8))) float v8f;

#define CC    64
#define HH    64
#define WW    64
#define DIN   192
#define DOUT  128
#define NEXP  8
#define TOKS  64          // tokens per workgroup = one image row (4 M-tiles)
#define TSTR  196         // padded token row stride (floats): bank-conflict-free
#define CCHUNK 8          // conv channels staged per chunk
#define PATCHN (CCHUNK * 3 * 66)          // 1584 floats
#define U1OFF  (TOKS * TSTR)              // 12544 : union1 = patch chunk | router_w
#define U2OFF  (U1OFF + PATCHN)           // 14128 : union2 = conv_w | logits+coeff
#define SMEMN  (U2OFF + DIN * 9)          // 15856 floats = 63424 B (< 64 KB)

__global__ __launch_bounds__(256) void moe_conv2d_fused(
    const float* __restrict__ x,
    const float* __restrict__ conv_w,
    const float* __restrict__ router_w,
    const float* __restrict__ w_experts,
    const float* __restrict__ b_experts,
    const float* __restrict__ w_shared,
    const float* __restrict__ b_shared,
    float* __restrict__ out)
{
    __shared__ float smem[SMEMN];
    float* s_tok = smem;          // 64 x 192 token tile (padded stride 196)
    float* s_u1  = smem + U1OFF;  // patch chunk (conv) | router weights (router)
    float* s_u2  = smem + U2OFF;  // conv weights (conv) | logits[512]+coeff (gemm)

    const int tid = threadIdx.x;
    const int b   = blockIdx.x >> 6;   // 16 images
    const int h   = blockIdx.x & 63;   // 64 rows

    // ---- stage depthwise filters into union2 ----
    for (int i = tid; i < DIN * 9; i += 256) s_u2[i] = conv_w[i];

    // ---- depthwise conv, 8 channels at a time ----
    for (int chunk = 0; chunk < CC / CCHUNK; ++chunk) {
        const int c0 = chunk * CCHUNK;
        __syncthreads();   // previous chunk fully consumed (also guards cw staging)
        // patch: [cl][r=0..2 -> h-1..h+1][col=0..65 -> w=-1..64], zero padded
        for (int i = tid; i < PATCHN; i += 256) {
            int cl  = i / 198;
            int rem = i - cl * 198;
            int r   = rem / 66;
            int col = rem - r * 66;
            int hs  = h + r - 1;
            int ws  = col - 1;
            float v = 0.0f;
            if ((unsigned)hs < (unsigned)HH && (unsigned)ws < (unsigned)WW)
                v = x[(((size_t)b * CC + c0 + cl) * HH + hs) * WW + ws];
            s_u1[i] = v;
        }
        __syncthreads();
        // features d in [c0*3, c0*3+24) for all 64 tokens (cross-correlation)
        for (int f = tid; f < TOKS * CCHUNK * 3; f += 256) {
            int tok = f / 24;
            int dd  = f - tok * 24;
            int d   = c0 * 3 + dd;
            int cl  = dd / 3;
            const float* wp = &s_u2[d * 9];
            const float* pp = &s_u1[(cl * 3) * 66 + tok];
            float acc = 0.0f;
            #pragma unroll
            for (int ky = 0; ky < 3; ++ky)
                #pragma unroll
                for (int kx = 0; kx < 3; ++kx)
                    acc = fmaf(wp[ky * 3 + kx], pp[ky * 66 + kx], acc);
            s_tok[tok * TSTR + d] = acc;
        }
    }
    __syncthreads();

    // ---- stage router weights into union1 (patch dead) ----
    for (int i = tid; i < DIN * NEXP; i += 256) s_u1[i] = router_w[i];
    __syncthreads();

    // ---- router logits (exact f32) into union2[0..512) (conv_w dead) ----
    for (int i = tid; i < TOKS * NEXP; i += 256) {
        int tok = i >> 3;
        int e   = i & 7;
        const float* tr = &s_tok[tok * TSTR];
        float acc = 0.0f;
        for (int k = 0; k < DIN; ++k)
            acc = fmaf(tr[k], s_u1[k * NEXP + e], acc);
        s_u2[i] = acc;
    }
    __syncthreads();

    // ---- top-2 gating; coeff[tok][e0..7, shared=8] at union2+512 ----
    float* s_coeff = s_u2 + TOKS * NEXP;
    if (tid < TOKS) {
        float l[NEXP];
        #pragma unroll
        for (int e = 0; e < NEXP; ++e) l[e] = s_u2[tid * NEXP + e];
        int i1 = 0; float m1 = l[0];
        #pragma unroll
        for (int e = 1; e < NEXP; ++e) if (l[e] > m1) { m1 = l[e]; i1 = e; }
        int i2 = (i1 == 0) ? 1 : 0; float m2 = l[i2];
        #pragma unroll
        for (int e = 0; e < NEXP; ++e)
            if (e != i1 && l[e] > m2) { m2 = l[e]; i2 = e; }
        float t   = expf(m2 - m1);
        float inv = 1.0f / (1.0f + t);
        #pragma unroll
        for (int e = 0; e < 9; ++e) s_coeff[tid * 12 + e] = 0.0f;
        s_coeff[tid * 12 + i1] = inv;
        s_coeff[tid * 12 + i2] = t * inv;
        s_coeff[tid * 12 + 8]  = 1.0f;
    }
    __syncthreads();

    // ---- GEMM: wave = 16-col tile; 4 M-tiles share each B fragment ----
    const int lane  = tid & 31;
    const int wv    = tid >> 5;
    const int half  = lane >> 4;
    const int nlane = lane & 15;
    const int n0    = wv << 4;

    const float* tr0 = &s_tok[(nlane     ) * TSTR + 2 * half];
    const float* tr1 = &s_tok[(nlane + 16) * TSTR + 2 * half];
    const float* tr2 = &s_tok[(nlane + 32) * TSTR + 2 * half];
    const float* tr3 = &s_tok[(nlane + 48) * TSTR + 2 * half];

    v8f fin0 = {0.f,0.f,0.f,0.f,0.f,0.f,0.f,0.f};
    v8f fin1 = fin0, fin2 = fin0, fin3 = fin0;

    for (int e = 0; e < 9; ++e) {
        const float* Wc = (e < 8) ? (w_experts + (size_t)e * DIN * DOUT) : w_shared;
        const float* bb = (e < 8) ? (b_experts + e * DOUT) : b_shared;
        float bn = bb[n0 + nlane];
        v8f acc0 = {bn,bn,bn,bn,bn,bn,bn,bn};   // bias folded into C init
        v8f acc1 = acc0, acc2 = acc0, acc3 = acc0;

        const float* bp = Wc + (2 * half) * DOUT + n0 + nlane;
        #pragma unroll 4
        for (int kk = 0; kk < 48; ++kk) {
            v2f bf;
            bf.x = bp[0];
            bf.y = bp[DOUT];
            bp += 4 * DOUT;
            v2f a0 = *(const v2f*)(tr0 + kk * 4);
            v2f a1 = *(const v2f*)(tr1 + kk * 4);
            v2f a2 = *(const v2f*)(tr2 + kk * 4);
            v2f a3 = *(const v2f*)(tr3 + kk * 4);
            acc0 = __builtin_amdgcn_wmma_f32_16x16x4_f32(false, a0, false, bf, (short)0, acc0, false, false);
            acc1 = __builtin_amdgcn_wmma_f32_16x16x4_f32(false, a1, false, bf, (short)0, acc1, false, false);
            acc2 = __builtin_amdgcn_wmma_f32_16x16x4_f32(false, a2, false, bf, (short)0, acc2, false, false);
            acc3 = __builtin_amdgcn_wmma_f32_16x16x4_f32(false, a3, false, bf, (short)0, acc3, false, false);
        }
        #pragma unroll
        for (int v = 0; v < 8; ++v) {
            int m = v + 8 * half;
            fin0[v] = fmaf(s_coeff[(m     ) * 12 + e], acc0[v], fin0[v]);
            fin1[v] = fmaf(s_coeff[(m + 16) * 12 + e], acc1[v], fin1[v]);
            fin2[v] = fmaf(s_coeff[(m + 32) * 12 + e], acc2[v], fin2[v]);
            fin3[v] = fmaf(s_coeff[(m + 48) * 12 + e], acc3[v], fin3[v]);
        }
    }

    // ---- store: lane owns one channel; 8 consecutive w per tile ----
    const int ch = n0 + nlane;
    float* op = out + (((size_t)b * DOUT + ch) * HH + h) * WW + 8 * half;
    v4f s;
    s = (v4f){fin0[0], fin0[1], fin0[2], fin0[3]}; *(v4f*)(op     ) = s;
    s = (v4f){fin0[4], fin0[5], fin0[6], fin0[7]}; *(v4f*)(op +  4) = s;
    s = (v4f){fin1[0], fin1[1], fin1[2], fin1[3]}; *(v4f*)(op + 16) = s;
    s = (v4f){fin1[4], fin1[5], fin1[6], fin1[7]}; *(v4f*)(op + 20) = s;
    s = (v4f){fin2[0], fin2[1], fin2[2], fin2[3]}; *(v4f*)(op + 32) = s;
    s = (v4f){fin2[4], fin2[5], fin2[6], fin2[7]}; *(v4f*)(op + 36) = s;
    s = (v4f){fin3[0], fin3[1], fin3[2], fin3[3]}; *(v4f*)(op + 48) = s;
    s = (v4f){fin3[4], fin3[5], fin3[6], fin3[7]}; *(v4f*)(op + 52) = s;
}

extern "C" void kernel_launch(void* const* d_in, const int* in_sizes, int n_in,
                              void* d_out, int out_size, void* d_ws, size_t ws_size,
                              hipStream_t stream) {
    const float* x         = (const float*)d_in[0];
    const float* conv_w    = (const float*)d_in[1];
    const float* router_w  = (const float*)d_in[2];
    const float* w_experts = (const float*)d_in[3];
    const float* b_experts = (const float*)d_in[4];
    const float* w_shared  = (const float*)d_in[5];
    const float* b_shared  = (const float*)d_in[6];
    float* out = (float*)d_out;
    (void)in_sizes; (void)n_in; (void)out_size; (void)d_ws; (void)ws_size;

    dim3 grid(16 * 64);   // one workgroup per (image, row) = 64 tokens
    dim3 block(256);      // 8 wave32 -> 8 column tiles of 16
    moe_conv2d_fused<<<grid, block, 0, stream>>>(
        x, conv_w, router_w, w_experts, b_experts, w_shared, b_shared, out);
}